// BerhuLoss_29678224015447
// MI455X (gfx1250) — compile-verified
//
#include <hip/hip_runtime.h>
#include <hip/hip_bf16.h>

typedef __attribute__((ext_vector_type(2))) float v2f;
typedef __attribute__((ext_vector_type(8))) float v8f;

#define BS 256          // threads per streaming block (8 wave32 waves)
#define WAVES_PER_BLOCK (BS / 32)

__device__ __forceinline__ float wave_max32(float v) {
#pragma unroll
    for (int m = 16; m >= 1; m >>= 1) v = fmaxf(v, __shfl_xor(v, m, 32));
    return v;
}

__device__ __forceinline__ float wave_sum32(float v) {
#pragma unroll
    for (int m = 16; m >= 1; m >>= 1) v += __shfl_xor(v, m, 32);
    return v;
}

// 32-lane sum where the first cross-lane step is done by V_WMMA_F32_16X16X4_F32.
// A = all-ones (16x4), B holds the per-lane partial in its first B-register and
// zero in the second, so every B element is covered exactly once:
//   D[m][n] = sum_k A[m][k]*B[k][n] = p_n + p_{n+16}   (for every m)
// C/D layout: c[0] at lane l carries column N = l%16, so c[0] = p_{l%16}+p_{l%16+16}.
// Four more xor-shuffles finish the 16-column sum. EXEC must be all ones.
__device__ __forceinline__ float wave_sum32_wmma(float p) {
    v2f a; a[0] = 1.0f; a[1] = 1.0f;
    v2f b; b[0] = p;    b[1] = 0.0f;
    v8f c = {};
    c = __builtin_amdgcn_wmma_f32_16x16x4_f32(false, a, false, b, (short)0, c,
                                              false, false);
    float v = c[0];
#pragma unroll
    for (int m = 8; m >= 1; m >>= 1) v += __shfl_xor(v, m, 32);
    return v;
}

// ---------------- Pass 1: per-block max of |target - input| ----------------
__global__ void berhu_max_pass(const float4* __restrict__ in4,
                               const float4* __restrict__ tg4,
                               const float* __restrict__ in1,
                               const float* __restrict__ tg1,
                               int n4, int n, float* __restrict__ block_max) {
    const int stride = gridDim.x * blockDim.x;
    int gid = blockIdx.x * blockDim.x + threadIdx.x;
    float m = 0.0f;
    for (int i = gid; i < n4; i += stride) {
        float4 x = in4[i], y = tg4[i];
        m = fmaxf(m, fabsf(y.x - x.x));
        m = fmaxf(m, fabsf(y.y - x.y));
        m = fmaxf(m, fabsf(y.z - x.z));
        m = fmaxf(m, fabsf(y.w - x.w));
    }
    for (int i = 4 * n4 + gid; i < n; i += stride)           // tail (none for this shape)
        m = fmaxf(m, fabsf(tg1[i] - in1[i]));

    m = wave_max32(m);
    __shared__ float s[WAVES_PER_BLOCK];
    const int lane = threadIdx.x & 31, w = threadIdx.x >> 5;
    if (lane == 0) s[w] = m;
    __syncthreads();
    if (w == 0) {
        m = (lane < WAVES_PER_BLOCK) ? s[lane] : 0.0f;
        m = wave_max32(m);
        if (lane == 0) block_max[blockIdx.x] = m;
    }
}

// ---------------- Pass 2: reduce block maxes -> c, 1/(2c) ----------------
__global__ void berhu_c_pass(const float* __restrict__ block_max, int nb,
                             float* __restrict__ cbuf) {
    // one block, 1024 threads
    float m = (threadIdx.x < (unsigned)nb) ? block_max[threadIdx.x] : 0.0f;
    m = wave_max32(m);
    __shared__ float s[32];
    const int lane = threadIdx.x & 31, w = threadIdx.x >> 5;
    if (lane == 0) s[w] = m;
    __syncthreads();
    if (w == 0) {
        m = (lane < (int)(blockDim.x >> 5)) ? s[lane] : 0.0f;
        m = wave_max32(m);
        if (lane == 0) {
            float c = 0.2f * m;
            cbuf[0] = c;
            cbuf[1] = (c > 0.0f) ? (0.5f / c) : 0.0f;   // 1/(2c); c==0 => all diffs 0
        }
    }
}

// ---------------- Pass 3: per-block BerHu sums ----------------
__global__ void berhu_sum_pass(const float4* __restrict__ in4,
                               const float4* __restrict__ tg4,
                               const float* __restrict__ in1,
                               const float* __restrict__ tg1,
                               int n4, int n, const float* __restrict__ cbuf,
                               float* __restrict__ block_sum) {
    const float c = cbuf[0];
    const float inv2c = cbuf[1];
    const float cc = c * c;
    const int stride = gridDim.x * blockDim.x;
    int gid = blockIdx.x * blockDim.x + threadIdx.x;
    float acc = 0.0f;
    for (int i = gid; i < n4; i += stride) {
        float4 x = in4[i], y = tg4[i];
        float d0 = fabsf(y.x - x.x);
        float d1 = fabsf(y.y - x.y);
        float d2 = fabsf(y.z - x.z);
        float d3 = fabsf(y.w - x.w);
        acc += (d0 <= c) ? d0 : (d0 * d0 + cc) * inv2c;
        acc += (d1 <= c) ? d1 : (d1 * d1 + cc) * inv2c;
        acc += (d2 <= c) ? d2 : (d2 * d2 + cc) * inv2c;
        acc += (d3 <= c) ? d3 : (d3 * d3 + cc) * inv2c;
    }
    for (int i = 4 * n4 + gid; i < n; i += stride) {
        float d = fabsf(tg1[i] - in1[i]);
        acc += (d <= c) ? d : (d * d + cc) * inv2c;
    }

    // wave reduction, first step via WMMA (all 256 threads active -> EXEC all 1s)
    acc = wave_sum32_wmma(acc);
    __shared__ float s[WAVES_PER_BLOCK];
    const int lane = threadIdx.x & 31, w = threadIdx.x >> 5;
    if (lane == 0) s[w] = acc;
    __syncthreads();
    if (w == 0) {
        acc = (lane < WAVES_PER_BLOCK) ? s[lane] : 0.0f;
        acc = wave_sum32(acc);
        if (lane == 0) block_sum[blockIdx.x] = acc;
    }
}

// ---------------- Pass 4: fixed-order final sum, write sum / batch ----------------
__global__ void berhu_final_pass(const float* __restrict__ block_sum, int nb,
                                 float inv_batch, float* __restrict__ out) {
    // one block, 1024 threads; reduction order fixed by code -> deterministic
    float v = (threadIdx.x < (unsigned)nb) ? block_sum[threadIdx.x] : 0.0f;
    v = wave_sum32(v);
    __shared__ float s[32];
    const int lane = threadIdx.x & 31, w = threadIdx.x >> 5;
    if (lane == 0) s[w] = v;
    __syncthreads();
    if (w == 0) {
        v = (lane < (int)(blockDim.x >> 5)) ? s[lane] : 0.0f;
        v = wave_sum32(v);
        if (lane == 0) out[0] = v * inv_batch;
    }
}

extern "C" void kernel_launch(void* const* d_in, const int* in_sizes, int n_in,
                              void* d_out, int out_size, void* d_ws, size_t ws_size,
                              hipStream_t stream) {
    const float* input  = (const float*)d_in[0];
    const float* target = (const float*)d_in[1];
    float* out = (float*)d_out;

    const int n  = in_sizes[0];          // 9,830,400
    const int n4 = n >> 2;

    // workspace layout: [0..nb) block maxes | [nb..2nb) block sums | [2nb..2nb+2) c,1/(2c)
    int nb = 1024;
    {   // clamp to available workspace (needs (2*nb+2)*4 bytes)
        long avail = (long)(ws_size / sizeof(float)) - 2;
        if (avail < 2) avail = 2;
        if ((long)nb * 2 > avail) nb = (int)(avail / 2);
        if (nb < 1) nb = 1;
    }
    float* block_max = (float*)d_ws;
    float* block_sum = block_max + nb;
    float* cbuf      = block_sum + nb;

    berhu_max_pass<<<nb, BS, 0, stream>>>((const float4*)input, (const float4*)target,
                                          input, target, n4, n, block_max);
    berhu_c_pass<<<1, 1024, 0, stream>>>(block_max, nb, cbuf);
    berhu_sum_pass<<<nb, BS, 0, stream>>>((const float4*)input, (const float4*)target,
                                          input, target, n4, n, cbuf, block_sum);
    berhu_final_pass<<<1, 1024, 0, stream>>>(block_sum, nb, 1.0f / 32.0f, out);
}